// VQVGAE_55774445306150
// MI455X (gfx1250) — compile-verified
//
#include <hip/hip_runtime.h>
#include <hip/hip_bf16.h>

// ---------------------------------------------------------------------------
// CDNA5 fp32 WMMA types and helpers
// ---------------------------------------------------------------------------
typedef float v2f __attribute__((ext_vector_type(2)));
typedef float v8f __attribute__((ext_vector_type(8)));

__device__ __forceinline__ v8f wmma_f32_4(v2f a, v2f b, v8f c) {
    // D = A(16x4) * B(4x16) + C(16x16), fp32 -> v_wmma_f32_16x16x4_f32
    return __builtin_amdgcn_wmma_f32_16x16x4_f32(
        /*neg_a=*/false, a, /*neg_b=*/false, b,
        /*c_mod=*/(short)0, c, /*reuse_a=*/false, /*reuse_b=*/false);
}

__device__ __forceinline__ float wave_reduce_add(float x) {
    #pragma unroll
    for (int off = 16; off > 0; off >>= 1) x += __shfl_xor(x, off, 32);
    return x;
}

// ---------------------------------------------------------------------------
// CDNA5 async global->LDS copy (ASYNCcnt-tracked, bypasses VGPRs).
// Guarded: fall back to plain load/store staging if toolchain lacks builtins.
// ---------------------------------------------------------------------------
#if __has_builtin(__builtin_amdgcn_global_load_async_to_lds_b128) && \
    __has_builtin(__builtin_amdgcn_s_wait_asynccnt)
#define HAVE_ASYNC_LDS 1
#else
#define HAVE_ASYNC_LDS 0
#endif

#if HAVE_ASYNC_LDS
// clang-22 prototypes the pointer params as (vector_size(16) int)* — match it.
typedef int v4i_b128 __attribute__((vector_size(16)));
typedef __attribute__((address_space(1))) v4i_b128* as1_v4i;
typedef __attribute__((address_space(3))) v4i_b128* as3_v4i;

__device__ __forceinline__ void async_copy16(const float* g, float* l) {
    // Generic->AS1: integer round trip. Generic->AS3: LDS aperture keeps the
    // byte offset in addr[31:0] (ISA 10.2), so truncation yields the DS offset.
    // AS-qualified args also implicitly widen to generic if the prototype is
    // generic.
    __builtin_amdgcn_global_load_async_to_lds_b128(
        (as1_v4i)(uintptr_t)g, (as3_v4i)(uint32_t)(uintptr_t)l,
        /*offset=*/0, /*cpol=*/0);
}
__device__ __forceinline__ void async_wait0() {
    __builtin_amdgcn_s_wait_asynccnt(0);
}
#endif

// Problem constants
#define BB 4096
#define NN 64
#define FF 64
#define HH 512
#define LL 256
#define KK 64
#define CC 3

// ---------------------------------------------------------------------------
// Fused GCN layer: Y = LN(A_b @ (X_b @ W) + bias) -> ReLU   (per-batch WG)
// K = input feature dim (64 for layer0, 512 for layer1).
// Dynamic LDS: As[64][68] | Xs[64][K+4] | Ts[64][68]
// ---------------------------------------------------------------------------
template <int K>
__global__ __launch_bounds__(256)
void gcn_layer_kernel(const float* __restrict__ adj,   // [B,64,64]
                      const float* __restrict__ Xin,   // [B,64,K]
                      const float* __restrict__ W,     // [K,512]
                      const float* __restrict__ bias,  // [512]
                      const float* __restrict__ gam,   // [512]
                      const float* __restrict__ bet,   // [512]
                      float* __restrict__ Y)           // [B,64,512]
{
    extern __shared__ __align__(16) float smem[];
    float* As = smem;                       // 64*68
    float* Xs = As + 64 * 68;               // 64*(K+4)
    float* Ts = Xs + 64 * (K + 4);          // 64*68

    const int b    = blockIdx.x;
    const int tid  = threadIdx.x;
    const int lane = tid & 31;
    const int w    = tid >> 5;

    const float* Ab = adj + (size_t)b * 64 * 64;
    const float* Xb = Xin + (size_t)b * 64 * K;

#if HAVE_ASYNC_LDS
    // ---- async stage A_b (64x64, pad 68) and X_b (64xK, pad K+4) ----
    #pragma unroll
    for (int i = tid; i < 1024; i += 256) {
        int r  = i >> 4;
        int c4 = (i & 15) * 4;
        async_copy16(Ab + r * 64 + c4, As + r * 68 + c4);
    }
    #pragma unroll 4
    for (int i = tid; i < 64 * K / 4; i += 256) {
        int r  = i / (K / 4);
        int c4 = (i % (K / 4)) * 4;
        async_copy16(Xb + r * K + c4, Xs + r * (K + 4) + c4);
    }
    async_wait0();
    __syncthreads();
#else
    #pragma unroll
    for (int i = tid; i < 1024; i += 256) {
        int r  = i >> 4;
        int c4 = (i & 15) * 4;
        float4 v = *(const float4*)(Ab + r * 64 + c4);
        *(float4*)(As + r * 68 + c4) = v;
    }
    #pragma unroll 4
    for (int i = tid; i < 64 * K / 4; i += 256) {
        int r  = i / (K / 4);
        int c4 = (i % (K / 4)) * 4;
        float4 v = *(const float4*)(Xb + r * K + c4);
        *(float4*)(Xs + r * (K + 4) + c4) = v;
    }
    __syncthreads();
#endif

    // subtile mapping: wave w handles (mi, ni0) and (mi, ni0+1)
    const int s0   = 2 * w;
    const int mi   = s0 >> 2;            // 0..3
    const int ni0  = s0 & 3;             // 0 or 2
    const int l15  = lane & 15;
    const int koff = (lane >> 4) * 2;    // A/B frag K sub-offset
    const int Mrow = mi * 16 + l15;      // A-frag row

    float* Yb = Y + (size_t)b * 64 * 512;

    for (int hc = 0; hc < 8; ++hc) {
        // ---- phase 1: T = Xs(64xK) @ W[:, hc*64 .. +64) ----
        v8f acc0 = {}; v8f acc1 = {};
        const int ncol0 = hc * 64 + ni0 * 16 + l15;
        #pragma unroll 4
        for (int kk = 0; kk < K; kk += 4) {
            v2f a;
            a.x = Xs[Mrow * (K + 4) + kk + koff];
            a.y = Xs[Mrow * (K + 4) + kk + koff + 1];
            const float* Wp = W + (size_t)(kk + koff) * 512;
            v2f b0; b0.x = Wp[ncol0];       b0.y = Wp[512 + ncol0];
            v2f b1; b1.x = Wp[ncol0 + 16];  b1.y = Wp[512 + ncol0 + 16];
            acc0 = wmma_f32_4(a, b0, acc0);
            acc1 = wmma_f32_4(a, b1, acc1);
        }
        __syncthreads();   // previous chunk's Ts consumers are done
        {
            const int rbase = mi * 16 + 8 * (lane >> 4);
            const int cb0   = ni0 * 16 + l15;
            #pragma unroll
            for (int v = 0; v < 8; ++v) {
                Ts[(rbase + v) * 68 + cb0]      = acc0[v];
                Ts[(rbase + v) * 68 + cb0 + 16] = acc1[v];
            }
        }
        __syncthreads();

        // ---- phase 2: Ychunk = As(64x64) @ Ts(64x64) + bias ----
        v8f y0 = {}; v8f y1 = {};
        const int cb0 = ni0 * 16 + l15;
        #pragma unroll
        for (int kk = 0; kk < 64; kk += 4) {
            v2f a;
            a.x = As[Mrow * 68 + kk + koff];
            a.y = As[Mrow * 68 + kk + koff + 1];
            v2f b0; b0.x = Ts[(kk + koff) * 68 + cb0];
                    b0.y = Ts[(kk + koff + 1) * 68 + cb0];
            v2f b1; b1.x = Ts[(kk + koff) * 68 + cb0 + 16];
                    b1.y = Ts[(kk + koff + 1) * 68 + cb0 + 16];
            y0 = wmma_f32_4(a, b0, y0);
            y1 = wmma_f32_4(a, b1, y1);
        }
        const float bb0 = bias[hc * 64 + ni0 * 16 + l15];
        const float bb1 = bias[hc * 64 + ni0 * 16 + 16 + l15];
        const int rbase = mi * 16 + 8 * (lane >> 4);
        const int cg0   = hc * 64 + ni0 * 16 + l15;
        #pragma unroll
        for (int v = 0; v < 8; ++v) {
            Yb[(rbase + v) * 512 + cg0]      = y0[v] + bb0;
            Yb[(rbase + v) * 512 + cg0 + 16] = y1[v] + bb1;
        }
    }
    __syncthreads();

    // ---- LayerNorm + ReLU (deterministic shuffle reduction), rows per wave
    for (int r = w; r < 64; r += 8) {
        float vals[16];
        float s = 0.f, ss = 0.f;
        #pragma unroll
        for (int j = 0; j < 16; ++j) {
            float x = Yb[r * 512 + lane + 32 * j];
            vals[j] = x; s += x; ss += x * x;
        }
        s  = wave_reduce_add(s);
        ss = wave_reduce_add(ss);
        const float mu   = s * (1.f / 512.f);
        const float var  = ss * (1.f / 512.f) - mu * mu;
        const float rstd = rsqrtf(var + 1e-5f);
        #pragma unroll
        for (int j = 0; j < 16; ++j) {
            const int cc = lane + 32 * j;
            float x = (vals[j] - mu) * rstd * gam[cc] + bet[cc];
            Yb[r * 512 + cc] = fmaxf(x, 0.f);
        }
    }
}

// ---------------------------------------------------------------------------
// Generic 64x64-tile fp32 WMMA GEMM: C = A[MxK] @ Bw[KxN] + bias[N]
// grid = (M/64, N/64), block = 256 (8 waves, 2 subtiles each).
// Double-buffered async global->LDS pipeline on CDNA5.
// ---------------------------------------------------------------------------
template <bool RELU>
__global__ __launch_bounds__(256)
void gemm64_kernel(const float* __restrict__ A,
                   const float* __restrict__ Bw,
                   const float* __restrict__ bias,
                   float* __restrict__ C,
                   int M, int K, int N)
{
    __shared__ __align__(16) float As[2][64 * 68];
    __shared__ __align__(16) float Bs[2][64 * 68];

    const int tid  = threadIdx.x;
    const int lane = tid & 31;
    const int w    = tid >> 5;
    const int m0   = blockIdx.x * 64;
    const int n0   = blockIdx.y * 64;

    const int s0   = 2 * w;
    const int mi   = s0 >> 2;
    const int ni0  = s0 & 3;
    const int l15  = lane & 15;
    const int koff = (lane >> 4) * 2;
    const int Mrow = mi * 16 + l15;
    const int cb0  = ni0 * 16 + l15;

    v8f acc0 = {}; v8f acc1 = {};

    auto compute_tile = [&](const float* Asb, const float* Bsb) {
        #pragma unroll
        for (int kk = 0; kk < 64; kk += 4) {
            v2f a;
            a.x = Asb[Mrow * 68 + kk + koff];
            a.y = Asb[Mrow * 68 + kk + koff + 1];
            v2f b0; b0.x = Bsb[(kk + koff) * 68 + cb0];
                    b0.y = Bsb[(kk + koff + 1) * 68 + cb0];
            v2f b1; b1.x = Bsb[(kk + koff) * 68 + cb0 + 16];
                    b1.y = Bsb[(kk + koff + 1) * 68 + cb0 + 16];
            acc0 = wmma_f32_4(a, b0, acc0);
            acc1 = wmma_f32_4(a, b1, acc1);
        }
    };

#if HAVE_ASYNC_LDS
    auto issue_tiles = [&](int k0, int bufi) {
        #pragma unroll
        for (int i = tid; i < 1024; i += 256) {
            int r  = i >> 4;
            int c4 = (i & 15) * 4;
            async_copy16(A + (size_t)(m0 + r) * K + k0 + c4, &As[bufi][r * 68 + c4]);
            async_copy16(Bw + (size_t)(k0 + r) * N + n0 + c4, &Bs[bufi][r * 68 + c4]);
        }
    };
    issue_tiles(0, 0);
    int buf = 0;
    for (int k0 = 0; k0 < K; k0 += 64) {
        async_wait0();          // my copies into `buf` have landed
        __syncthreads();        // everyone's landed; prev reads of buf^1 done
        if (k0 + 64 < K) issue_tiles(k0 + 64, buf ^ 1);  // overlaps compute
        compute_tile(&As[buf][0], &Bs[buf][0]);
        buf ^= 1;
    }
#else
    for (int k0 = 0; k0 < K; k0 += 64) {
        __syncthreads();
        #pragma unroll
        for (int i = tid; i < 1024; i += 256) {
            int r  = i >> 4;
            int c4 = (i & 15) * 4;
            float4 va = *(const float4*)(A + (size_t)(m0 + r) * K + k0 + c4);
            *(float4*)(&As[0][r * 68 + c4]) = va;
            float4 vb = *(const float4*)(Bw + (size_t)(k0 + r) * N + n0 + c4);
            *(float4*)(&Bs[0][r * 68 + c4]) = vb;
        }
        __syncthreads();
        compute_tile(&As[0][0], &Bs[0][0]);
    }
#endif

    const int rbase = mi * 16 + 8 * (lane >> 4);
    const int c0    = n0 + ni0 * 16 + l15;
    const float bb0 = bias[c0];
    const float bb1 = bias[c0 + 16];
    #pragma unroll
    for (int v = 0; v < 8; ++v) {
        float o0 = acc0[v] + bb0;
        float o1 = acc1[v] + bb1;
        if (RELU) { o0 = fmaxf(o0, 0.f); o1 = fmaxf(o1, 0.f); }
        C[(size_t)(m0 + rbase + v) * N + c0]      = o0;
        C[(size_t)(m0 + rbase + v) * N + c0 + 16] = o1;
    }
}

// ---------------------------------------------------------------------------
// VQ distances + argmin + one-hot, and condition regressor (1 wave / row)
// ---------------------------------------------------------------------------
__global__ __launch_bounds__(256)
void vq_head_kernel(const float* __restrict__ z,       // [4096,256]
                    const float* __restrict__ hpre,    // [4096,256] (pre-ReLU)
                    const float* __restrict__ codebook,// [64,256]
                    const float* __restrict__ crw2,    // [256,3]
                    const float* __restrict__ crb2,    // [3]
                    float* __restrict__ pred,          // [4096,3]
                    float* __restrict__ enc)           // [4096,64]
{
    const int lane = threadIdx.x & 31;
    const int w    = threadIdx.x >> 5;
    const int r    = blockIdx.x * 8 + w;

    // load z row, 8 elements per lane
    float zv[8];
    float zz = 0.f;
    #pragma unroll
    for (int j = 0; j < 8; ++j) {
        zv[j] = z[(size_t)r * 256 + lane + 32 * j];
        zz += zv[j] * zv[j];
    }
    zz = wave_reduce_add(zz);

    float best = 3.4e38f;
    int   bidx = 0;
    for (int k = 0; k < 64; ++k) {
        float dot = 0.f, cn = 0.f;
        #pragma unroll
        for (int j = 0; j < 8; ++j) {
            float c = codebook[k * 256 + lane + 32 * j];
            dot += zv[j] * c;
            cn  += c * c;
        }
        dot = wave_reduce_add(dot);
        cn  = wave_reduce_add(cn);
        float dist = zz + cn - 2.f * dot;
        if (dist < best) { best = dist; bidx = k; }   // first-min, like argmin
    }
    #pragma unroll
    for (int k = lane; k < 64; k += 32)
        enc[(size_t)r * 64 + k] = (k == bidx) ? 1.f : 0.f;

    // regressor: pred = relu(relu(h) @ crw2 + crb2)
    float hv[8];
    #pragma unroll
    for (int j = 0; j < 8; ++j)
        hv[j] = fmaxf(hpre[(size_t)r * 256 + lane + 32 * j], 0.f);
    #pragma unroll
    for (int c = 0; c < 3; ++c) {
        float acc = 0.f;
        #pragma unroll
        for (int j = 0; j < 8; ++j)
            acc += hv[j] * crw2[(lane + 32 * j) * 3 + c];
        acc = wave_reduce_add(acc);
        if (lane == 0) pred[(size_t)r * 3 + c] = fmaxf(acc + crb2[c], 0.f);
    }
}

// ---------------------------------------------------------------------------
// Launcher
// ---------------------------------------------------------------------------
extern "C" void kernel_launch(void* const* d_in, const int* in_sizes, int n_in,
                              void* d_out, int out_size, void* d_ws, size_t ws_size,
                              hipStream_t stream)
{
    const float* adj   = (const float*)d_in[0];
    const float* feats = (const float*)d_in[1];
    const float* W0    = (const float*)d_in[2];
    const float* b0    = (const float*)d_in[3];
    const float* g0    = (const float*)d_in[4];
    const float* be0   = (const float*)d_in[5];
    const float* W1    = (const float*)d_in[6];
    const float* b1    = (const float*)d_in[7];
    const float* g1    = (const float*)d_in[8];
    const float* be1   = (const float*)d_in[9];
    const float* fc_w  = (const float*)d_in[10];
    const float* fc_b  = (const float*)d_in[11];
    const float* cbk   = (const float*)d_in[12];
    const float* crw1  = (const float*)d_in[13];
    const float* crb1  = (const float*)d_in[14];
    const float* crw2  = (const float*)d_in[15];
    const float* crb2  = (const float*)d_in[16];

    float* out_z    = (float*)d_out;                    // [4096,256]
    float* out_pred = out_z + (size_t)BB * LL;          // [4096,3]
    float* out_enc  = out_pred + (size_t)BB * CC;       // [4096,64]

    float* xbuf = (float*)d_ws;                         // [B,64,512] (reused in place)
    float* hbuf = xbuf + (size_t)BB * NN * HH;          // [4096,256]

    // Layer 0: features(64) -> hidden(512)
    {
        const size_t smem = (size_t)(64 * 68 + 64 * (64 + 4) + 64 * 68) * sizeof(float);
        gcn_layer_kernel<64><<<BB, 256, smem, stream>>>(adj, feats, W0, b0, g0, be0, xbuf);
    }
    // Layer 1: hidden(512) -> hidden(512), in place on xbuf
    {
        const size_t smem = (size_t)(64 * 68 + 64 * (512 + 4) + 64 * 68) * sizeof(float);
        gcn_layer_kernel<512><<<BB, 256, smem, stream>>>(adj, xbuf, W1, b1, g1, be1, xbuf);
    }
    // z = xflat(4096 x 32768) @ fc_w(32768 x 256) + fc_b
    gemm64_kernel<false><<<dim3(BB / 64, LL / 64), 256, 0, stream>>>(
        xbuf, fc_w, fc_b, out_z, BB, NN * HH, LL);
    // h_pre = z @ cr_w1 + cr_b1  (ReLU applied on read in the head kernel)
    gemm64_kernel<false><<<dim3(BB / 64, LL / 64), 256, 0, stream>>>(
        out_z, crw1, crb1, hbuf, BB, LL, LL);
    // VQ argmin/one-hot + regressor
    vq_head_kernel<<<BB / 8, 256, 0, stream>>>(out_z, hbuf, cbk, crw2, crb2,
                                               out_pred, out_enc);
}